// Model_89704686944625
// MI455X (gfx1250) — compile-verified
//
#include <hip/hip_runtime.h>
#include <stdint.h>

// Cloth Verlet step, G x G grid of float3 points.
// CDNA5 path: TDM (tensor_load_to_lds) stages tile+halo into LDS, tracked by
// TENSORcnt; compute waves read neighbors from LDS (conflict-free 3-dword
// stride), integrate, and store float3 results with b96-class accesses.

#define G     2048
#define TW    64              // tile width  (points)
#define TH    4               // tile height (points)
#define HW    (TW + 2)        // halo width  = 66
#define HH    (TH + 2)        // halo height = 6
#define ROWF  (HW * 3)        // LDS floats per halo row = 198

typedef uint32_t u32;
typedef uint64_t u64;
typedef u32 u32x4 __attribute__((ext_vector_type(4)));
typedef int i32x4 __attribute__((ext_vector_type(4)));
typedef int i32x8 __attribute__((ext_vector_type(8)));

__global__ __launch_bounds__(256) void cloth_tdm_kernel(
    const float* __restrict__ pos,
    const float* __restrict__ prev,
    float* __restrict__ out) {
  __shared__ float tile[HH * ROWF];   // 4752 B of the 320 KB WGP LDS

  const int i0 = blockIdx.y * TH;     // tile origin (row)
  const int j0 = blockIdx.x * TW;     // tile origin (col)

  // Clamped halo window [rs..re] x [cs..ce]; cells outside the grid stay
  // unloaded in LDS and are excluded by selects below (never multiplied).
  const int rs = (i0 > 0) ? (i0 - 1) : 0;
  const int re = (i0 + TH < G) ? (i0 + TH) : (G - 1);
  const int cs = (j0 > 0) ? (j0 - 1) : 0;
  const int ce = (j0 + TW < G) ? (j0 + TW) : (G - 1);
  const int nrows  = re - rs + 1;            // TH+1 or TH+2
  const int widthE = (ce - cs + 1) * 3;      // floats per loaded row (<= 198)
  const int ldsRow0 = rs - (i0 - 1);         // 0 or 1 (top clamp shift)
  const int ldsCol0 = cs - (j0 - 1);         // 0 or 1 (left clamp shift)

  if (threadIdx.x < 32u) {  // wave 0 issues the tensor DMA for the workgroup
    const u64 gaddr = (u64)(uintptr_t)pos + ((u64)rs * G + (u64)cs) * 12u;
    const u32 laddr = (u32)(uintptr_t)(&tile[0]) +
                      (u32)(ldsRow0 * ROWF + ldsCol0 * 3) * 4u;

    // ---- D# group 0 (ISA 08 Async/Tensor sec 8.3) ----
    u32x4 g0;
    g0.x = 1u;                                        // count=1, user mode
    g0.y = laddr;                                     // lds_addr (bytes)
    g0.z = (u32)(gaddr & 0xFFFFFFFFu);                // global_addr[31:0]
    g0.w = (u32)((gaddr >> 32) & 0x1FFFFFFu)          // global_addr[56:32]
         | (2u << 30);                                // type=2 ("image")

    // ---- D# group 1 (sec 8.4): 1-row tiles, iterate mode ----
    const u32 tdim = 0x7FFFFFFFu;                     // huge dims: no OOB clip
    i32x8 g1;
    g1[0] = (int)((2u << 16) | (1u << 19));           // data_size=4B, iterate_en
    g1[1] = (int)((tdim & 0xFFFFu) << 16);            // tensor_dim0[15:0]
    g1[2] = (int)((tdim >> 16) | ((tdim & 0xFFFFu) << 16)); // dim0 hi | dim1 lo
    g1[3] = (int)((tdim >> 16) | ((u32)widthE << 16));      // dim1 hi | tile_dim0
    g1[4] = (int)1;                                   // tile_dim1=1, tile_dim2=0
    g1[5] = (int)(G * 3);                             // tensor_dim0_stride lo32
    g1[6] = 0;                                        // stride0 hi | stride1 lo
    g1[7] = 0;                                        // stride1 hi

    // ---- D# group 2 (sec 8.5): iteration control ----
    i32x4 g2;
    g2[0] = 0;                                        // tensor_dim2 (unused)
    g2[1] = ROWF;                                     // lds_addr_increment/iter
    g2[2] = G * 3;                                    // global_addr_increment lo
    g2[3] = (int)((u32)(nrows - 1) << 16);            // iterate_count (0 => 1x)

    i32x4 g3 = {0, 0, 0, 0};                          // group 3 unused

#if __clang_major__ >= 23
    i32x8 gx = {0, 0, 0, 0, 0, 0, 0, 0};
    __builtin_amdgcn_tensor_load_to_lds(g0, g1, g2, g3, gx, 0);
#else
    __builtin_amdgcn_tensor_load_to_lds(g0, g1, g2, g3, 0);
#endif
    __builtin_amdgcn_s_wait_tensorcnt(0);             // DMA complete for wave 0
  }
  __syncthreads();                                    // publish LDS tile

  // ---- per-point stencil from LDS ----
  const int c = (int)(threadIdx.x & (TW - 1));
  const int r = (int)(threadIdx.x >> 6);
  const int i = i0 + r;
  const int j = j0 + c;

  const float* lp = &tile[(r + 1) * ROWF + (c + 1) * 3];
  const float px = lp[0], py = lp[1], pz = lp[2];

  float fx = 0.0f, fy = -9.81f, fz = 0.0f;            // gravity
  const int di[4] = {0, 0, 1, -1};
  const int dj[4] = {1, -1, 0, 0};
#pragma unroll
  for (int t = 0; t < 4; ++t) {
    const int ni = i + di[t];
    const int nj = j + dj[t];
    const bool m = ((unsigned)ni < (unsigned)G) && ((unsigned)nj < (unsigned)G);
    const float* np = &tile[(r + 1 + di[t]) * ROWF + (c + 1 + dj[t]) * 3];
    const float dx = np[0] - px;
    const float dy = np[1] - py;
    const float dz = np[2] - pz;
    const float d2   = dx * dx + dy * dy + dz * dz + 1e-8f;
    const float dist = sqrtf(d2);
    const float coef = 500.0f * (dist - 1.0f) / dist;
    // select (not multiply): halo garbage must never reach the accumulator
    fx += m ? coef * dx : 0.0f;
    fy += m ? coef * dy : 0.0f;
    fz += m ? coef * dz : 0.0f;
  }

  const u64 idx = ((u64)i * G + (u64)j) * 3u;
  const float qx = prev[idx + 0];
  const float qy = prev[idx + 1];
  const float qz = prev[idx + 2];
  const float dt2 = 0.001f * 0.001f;
  out[idx + 0] = 2.0f * px - qx + fx * dt2;
  out[idx + 1] = 2.0f * py - qy + fy * dt2;
  out[idx + 2] = 2.0f * pz - qz + fz * dt2;
}

extern "C" void kernel_launch(void* const* d_in, const int* in_sizes, int n_in,
                              void* d_out, int out_size, void* d_ws, size_t ws_size,
                              hipStream_t stream) {
  (void)in_sizes; (void)n_in; (void)out_size; (void)d_ws; (void)ws_size;
  const float* pos  = (const float*)d_in[0];
  const float* prev = (const float*)d_in[1];
  float*       out  = (float*)d_out;
  dim3 grid(G / TW, G / TH);   // 32 x 512 = 16384 blocks
  cloth_tdm_kernel<<<grid, 256, 0, stream>>>(pos, prev, out);
}